// GCN_74345883894179
// MI455X (gfx1250) — compile-verified
//
#include <hip/hip_runtime.h>

#define NN 50000
#define NE 800000
#define F  128
#define FP 132   // LDS row stride (pad 4 words -> conflict-free banks)
#define FC 40
#define FCP 48   // padded classifier width (3 x 16)
#define SCAN_T 1024

typedef float v2f __attribute__((ext_vector_type(2)));
typedef float v8f __attribute__((ext_vector_type(8)));

// ---------------- CSR build: count ----------------
__global__ void k_zero_cnt(int* cnt) {
    int i = blockIdx.x * blockDim.x + threadIdx.x;
    if (i < NN) cnt[i] = 0;
}

__global__ void k_count(const long long* __restrict__ dst, int* cnt) {
    int e = blockIdx.x * blockDim.x + threadIdx.x;
    if (e < NE) atomicAdd(&cnt[(int)dst[e]], 1);
}

// single-block exclusive scan: rowptr[0..NN], rowptr[NN] = E
__global__ __launch_bounds__(SCAN_T) void k_scan(const int* __restrict__ cnt,
                                                 int* __restrict__ rowptr) {
    __shared__ int ssum[SCAN_T];
    const int t = threadIdx.x;
    const int chunk = (NN + SCAN_T - 1) / SCAN_T;   // 49
    int beg = t * chunk;
    int end = min(beg + chunk, NN);
    int s = 0;
    for (int i = beg; i < end; ++i) s += cnt[i];
    ssum[t] = s;
    __syncthreads();
    for (int off = 1; off < SCAN_T; off <<= 1) {    // inclusive Hillis-Steele
        int v = (t >= off) ? ssum[t - off] : 0;
        __syncthreads();
        ssum[t] += v;
        __syncthreads();
    }
    int base = (t == 0) ? 0 : ssum[t - 1];
    for (int i = beg; i < end; ++i) {
        rowptr[i] = base;
        base += cnt[i];
    }
    if (t == SCAN_T - 1) rowptr[NN] = ssum[SCAN_T - 1];
}

// next[] = rowptr copy (fill cursor); dinv = rsqrt(deg + 1 self-loop)
__global__ void k_prep(const int* __restrict__ rowptr, const int* __restrict__ cnt,
                       int* __restrict__ next, float* __restrict__ dinv) {
    int i = blockIdx.x * blockDim.x + threadIdx.x;
    if (i >= NN) return;
    next[i] = rowptr[i];
    dinv[i] = rsqrtf((float)cnt[i] + 1.0f);
}

__global__ void k_fill(const long long* __restrict__ src, const long long* __restrict__ dst,
                       int* __restrict__ next, int* __restrict__ colbuf) {
    int e = blockIdx.x * blockDim.x + threadIdx.x;
    if (e >= NE) return;
    int d = (int)dst[e];
    int pos = atomicAdd(&next[d], 1);
    colbuf[pos] = (int)src[e];
}

// ---------------- GEMM: P = (X @ W) * dinv[row], LDS-staged A tile ----------
// block = 256 threads = 8 waves; wave w -> 16x16 tile at cols [16w, 16w+16)
__global__ __launch_bounds__(256) void k_gemm128(
    const float* __restrict__ X, const float* __restrict__ W,
    const float* __restrict__ dinv, float* __restrict__ P) {
    __shared__ float As[16 * FP];            // 16 rows x 128 (+4 pad) = 8.25 KB
    const int tid = threadIdx.x;
    const int row_tile = blockIdx.x;         // 0..3124
    const float* Xblk = X + (size_t)row_tile * 16 * F;
    #pragma unroll
    for (int i = tid; i < 16 * F; i += 256) {
        int r = i >> 7, cc = i & (F - 1);
        As[r * FP + cc] = Xblk[i];           // coalesced global read
    }
    __syncthreads();

    const int wave = tid >> 5;
    const int lane = tid & 31;
    const int half = lane >> 4;              // 0: K pair {0,1}; 1: {2,3}
    const int sub  = lane & 15;
    const int col0 = wave * 16;

    const float* Arow = As + sub * FP;       // A: M = sub (LDS)
    const float* Bcol = W + col0 + sub;      // B: N = sub (L2-resident)

    v8f c = {};
    #pragma unroll 8
    for (int k = 0; k < F; k += 4) {
        v2f a, b;
        a.x = Arow[k + 2 * half];
        a.y = Arow[k + 2 * half + 1];
        b.x = Bcol[(size_t)(k + 2 * half) * F];
        b.y = Bcol[(size_t)(k + 2 * half + 1) * F];
        c = __builtin_amdgcn_wmma_f32_16x16x4_f32(
                false, a, false, b, (short)0, c, false, false);
    }
    #pragma unroll
    for (int r = 0; r < 8; ++r) {
        int m   = r + 8 * half;
        int row = row_tile * 16 + m;
        P[(size_t)row * F + col0 + sub] = c[r] * dinv[row];
    }
}

// ---------------- gather-aggregate + finalize --------------------------------
// wave per node: acc = P[n] (self) + sum_{nbr} P[nbr]; out = relu(dinv*acc + b)
__global__ __launch_bounds__(256) void k_aggregate(
    const int* __restrict__ rowptr, const int* __restrict__ colbuf,
    const float* __restrict__ P, const float* __restrict__ dinv,
    const float* __restrict__ bias, float* __restrict__ out) {
    int n = blockIdx.x * 8 + (threadIdx.x >> 5);
    if (n >= NN) return;
    int lane = threadIdx.x & 31;
    int beg = rowptr[n], end = rowptr[n + 1];

    float4 acc = *(const float4*)(P + (size_t)n * F + lane * 4);   // self term
    for (int j = beg; j < end; j += 32) {
        int myidx = (j + lane < end) ? colbuf[j + lane] : 0;       // coalesced
        int cnt32 = min(32, end - j);
        for (int t = 0; t < cnt32; ++t) {
            int sid = __shfl(myidx, t);
            const float4 v = *(const float4*)(P + (size_t)sid * F + lane * 4);
            acc.x += v.x; acc.y += v.y; acc.z += v.z; acc.w += v.w;
        }
    }
    float dv = dinv[n];
    int c0 = lane * 4;
    float4 r;
    r.x = fmaxf(dv * acc.x + bias[c0 + 0], 0.0f);
    r.y = fmaxf(dv * acc.y + bias[c0 + 1], 0.0f);
    r.z = fmaxf(dv * acc.z + bias[c0 + 2], 0.0f);
    r.w = fmaxf(dv * acc.w + bias[c0 + 3], 0.0f);
    *(float4*)(out + (size_t)n * F + c0) = r;
}

// ---------------- pad classifier weights 128x40 -> 128x48 ----------------
__global__ void k_pad_wc(const float* __restrict__ Wc, float* __restrict__ Wp) {
    int i = blockIdx.x * blockDim.x + threadIdx.x;
    if (i >= F * FCP) return;
    int k = i / FCP, n = i % FCP;
    Wp[i] = (n < FC) ? Wc[k * FC + n] : 0.0f;
}

// ---------------- classifier GEMM: out = X @ WcPad + bc (store cols < 40) ----
__global__ __launch_bounds__(256) void k_gemm_cls(
    const float* __restrict__ X, const float* __restrict__ Wp,
    const float* __restrict__ bc, float* __restrict__ out) {
    const int wave_g = blockIdx.x * 8 + (threadIdx.x >> 5);
    const int total  = 3125 * 3;             // row tiles x col tiles
    if (wave_g >= total) return;             // whole-wave uniform exit
    const int lane = threadIdx.x & 31;
    const int half = lane >> 4;
    const int sub  = lane & 15;
    const int row_tile = wave_g / 3;
    const int col_tile = wave_g % 3;
    const int col0 = col_tile * 16;

    const float* Arow = X  + (size_t)(row_tile * 16 + sub) * F;
    const float* Bcol = Wp + col0 + sub;

    v8f c = {};
    #pragma unroll 4
    for (int k = 0; k < F; k += 4) {
        v2f a, b;
        a.x = Arow[k + 2 * half];
        a.y = Arow[k + 2 * half + 1];
        b.x = Bcol[(size_t)(k + 2 * half) * FCP];
        b.y = Bcol[(size_t)(k + 2 * half + 1) * FCP];
        c = __builtin_amdgcn_wmma_f32_16x16x4_f32(
                false, a, false, b, (short)0, c, false, false);
    }
    int col = col0 + sub;
    #pragma unroll
    for (int r = 0; r < 8; ++r) {
        int m   = r + 8 * half;
        int row = row_tile * 16 + m;
        if (col < FC) out[(size_t)row * FC + col] = c[r] + bc[col];
    }
}

static inline size_t align256(size_t x) { return (x + 255) & ~(size_t)255; }

extern "C" void kernel_launch(void* const* d_in, const int* in_sizes, int n_in,
                              void* d_out, int out_size, void* d_ws, size_t ws_size,
                              hipStream_t stream) {
    const float*     x   = (const float*)d_in[0];
    const long long* ei  = (const long long*)d_in[1];   // int64 [2, E]
    const float*     W1  = (const float*)d_in[2];
    const float*     b1  = (const float*)d_in[3];
    const float*     W2  = (const float*)d_in[4];
    const float*     b2  = (const float*)d_in[5];
    const float*     Wc  = (const float*)d_in[6];
    const float*     bc  = (const float*)d_in[7];
    float*           out = (float*)d_out;

    const long long* src = ei;
    const long long* dst = ei + NE;

    // workspace layout (rebuilt from scratch every call)
    char* ws = (char*)d_ws;
    size_t o = 0;
    int*   cnt    = (int*)  (ws + o); o += align256((size_t)NN * 4);
    int*   rowptr = (int*)  (ws + o); o += align256((size_t)(NN + 1) * 4);
    int*   nxt    = (int*)  (ws + o); o += align256((size_t)NN * 4);
    int*   colbuf = (int*)  (ws + o); o += align256((size_t)NE * 4);
    float* dinv   = (float*)(ws + o); o += align256((size_t)NN * 4);
    float* P      = (float*)(ws + o); o += align256((size_t)NN * F * 4);
    float* Xb     = (float*)(ws + o); o += align256((size_t)NN * F * 4);
    float* WcP    = (float*)(ws + o); o += align256((size_t)F * FCP * 4);
    (void)ws_size; (void)in_sizes; (void)n_in; (void)out_size;

    // CSR build + dinv
    k_zero_cnt<<<(NN + 255) / 256, 256, 0, stream>>>(cnt);
    k_count   <<<(NE + 255) / 256, 256, 0, stream>>>(dst, cnt);
    k_scan    <<<1, SCAN_T, 0, stream>>>(cnt, rowptr);
    k_prep    <<<(NN + 255) / 256, 256, 0, stream>>>(rowptr, cnt, nxt, dinv);
    k_fill    <<<(NE + 255) / 256, 256, 0, stream>>>(src, dst, nxt, colbuf);

    // layer 1
    k_gemm128  <<<NN / 16, 256, 0, stream>>>(x, W1, dinv, P);
    k_aggregate<<<(NN + 7) / 8, 256, 0, stream>>>(rowptr, colbuf, P, dinv, b1, Xb);

    // layer 2
    k_gemm128  <<<NN / 16, 256, 0, stream>>>(Xb, W2, dinv, P);
    k_aggregate<<<(NN + 7) / 8, 256, 0, stream>>>(rowptr, colbuf, P, dinv, b2, Xb);

    // classifier
    k_pad_wc  <<<(F * FCP + 255) / 256, 256, 0, stream>>>(Wc, WcP);
    k_gemm_cls<<<(3125 * 3 + 7) / 8, 256, 0, stream>>>(Xb, WcP, bc, out);
}